// ImageTransformer_36490042147683
// MI455X (gfx1250) — compile-verified
//
#include <hip/hip_runtime.h>

// Fused STN homography warp + bilinear sample + grayscale + patch gather.
// B=256, IMG 240x320x3 f32, P=128. One thread per output pixel; block = one
// (batch, 256-pixel tile); H_eff per batch computed into LDS once per block.
// Projective transform T = H_eff * [gx,gy,1] is done with
// V_WMMA_F32_16X16X4_F32 (two WMMAs per wave cover all 32 lane-pixels).

typedef float v2f __attribute__((ext_vector_type(2)));
typedef float v8f __attribute__((ext_vector_type(8)));

#define IMG_H 240
#define IMG_W 320
#define PATCH 128

__global__ __launch_bounds__(256) void warp_gather_kernel(
    const float* __restrict__ I,       // [B,240,320,3]
    const float* __restrict__ H_mat,   // [B,3,3]
    const float* __restrict__ Mm,      // [3,3]
    const float* __restrict__ Mi,      // [3,3]
    const int*   __restrict__ pidx,    // [B,128,128]
    float* __restrict__ out)           // [B,128,128]
{
    __shared__ float sH[9];
    const int b = blockIdx.y;
    const int t = blockIdx.x * 256 + threadIdx.x;

    // ---- H_eff = M_inv @ H_mat[b] @ M (threads 0..8) ----
    if (threadIdx.x < 9) {
        const int r = threadIdx.x / 3, c = threadIdx.x % 3;
        const float* Hb = H_mat + b * 9;
        float acc = 0.f;
#pragma unroll
        for (int j = 0; j < 3; ++j) {
            float hm = 0.f;
#pragma unroll
            for (int k = 0; k < 3; ++k) hm += Hb[j * 3 + k] * Mm[k * 3 + c];
            acc += Mi[r * 3 + j] * hm;
        }
        sH[threadIdx.x] = acc;
    }
    __syncthreads();

    // ---- per-lane pixel position from fused gather index ----
    const int idx = pidx[b * (PATCH * PATCH) + t];
    const int py = idx / IMG_W;
    const int px = idx - py * IMG_W;
    const float gx = 2.f * (float)px * (1.f / (float)(IMG_W - 1)) - 1.f;
    const float gy = 2.f * (float)py * (1.f / (float)(IMG_H - 1)) - 1.f;

    const unsigned lane = threadIdx.x & 31u;
    const bool lo = lane < 16u;

    // ---- A operand (16x4 f32): rows 0..2 and 8..10 = H_eff rows, col3 = 0.
    // ISA layout: lanes 0-15 hold K=0 (v0), K=1 (v1); lanes 16-31 hold K=2,K=3.
    const int m  = (int)(lane & 15u);
    const int hr = (m < 3) ? m : ((m >= 8 && m < 11) ? (m - 8) : -1);
    const int hc = (hr >= 0) ? hr : 0;
    const float hv0 = (hr >= 0) ? sH[hc * 3 + 0] : 0.f;
    const float hv1 = (hr >= 0) ? sH[hc * 3 + 1] : 0.f;
    const float hv2 = (hr >= 0) ? sH[hc * 3 + 2] : 0.f;
    v2f A;
    A.x = lo ? hv0 : hv2;   // K=0 / K=2
    A.y = lo ? hv1 : 0.f;   // K=1 / K=3 (zero column)

    // ---- B operands (4x16 f32): col n = [gx, gy, 1, 0] of a pixel.
    // Layout: v0 lanes 0-15 = row K=0, lanes 16-31 = row K=2; v1 = K=1 / K=3.
    const float gxh = __shfl(gx, (int)(lane & 15u) + 16, 32);
    const float gyh = __shfl(gy, (int)(lane & 15u) + 16, 32);
    v2f B0, B1;
    B0.x = lo ? gx  : 1.f;  B0.y = lo ? gy  : 0.f;  // pixels of lanes 0..15
    B1.x = lo ? gxh : 1.f;  B1.y = lo ? gyh : 0.f;  // pixels of lanes 16..31

    v8f Cz = {};
    v8f D0 = __builtin_amdgcn_wmma_f32_16x16x4_f32(
        false, A, false, B0, (short)0, Cz, false, false);
    v8f D1 = __builtin_amdgcn_wmma_f32_16x16x4_f32(
        false, A, false, B1, (short)0, Cz, false, false);

    // lane n<16:  D0 vgpr r = D[r][n]      -> rows 0..2 = T of own pixel
    // lane 16+n:  D1 vgpr r = D[r+8][n]    -> rows 8..10 = T of own pixel
    const float Tx = lo ? D0[0] : D1[0];
    const float Ty = lo ? D0[1] : D1[1];
    const float Tz = lo ? D0[2] : D1[2];

    // ---- perspective divide (reference's clamp semantics) ----
    const float den = (fabsf(Tz) < 1e-7f) ? 1e-7f : Tz;
    const float inv = 1.f / den;
    const float xs = (Tx * inv + 1.f) * ((float)IMG_W * 0.5f);
    const float ys = (Ty * inv + 1.f) * ((float)IMG_H * 0.5f);

    // ---- edge-clipped bilinear sample + grayscale ----
    const int fx = (int)floorf(xs);
    const int fy = (int)floorf(ys);
    const int x0 = min(max(fx,     0), IMG_W - 1);
    const int x1 = min(max(fx + 1, 0), IMG_W - 1);
    const int y0 = min(max(fy,     0), IMG_H - 1);
    const int y1 = min(max(fy + 1, 0), IMG_H - 1);

    const float x0f = (float)x0, x1f = (float)x1;
    const float y0f = (float)y0, y1f = (float)y1;
    const float wa = (x1f - xs) * (y1f - ys);
    const float wb = (x1f - xs) * (ys - y0f);
    const float wc = (xs - x0f) * (y1f - ys);
    const float wd = (xs - x0f) * (ys - y0f);

    const float* img = I + (size_t)b * (IMG_H * IMG_W * 3);
    const int r00 = (y0 * IMG_W + x0) * 3;
    const int r10 = (y1 * IMG_W + x0) * 3;
    const int r01 = (y0 * IMG_W + x1) * 3;
    const int r11 = (y1 * IMG_W + x1) * 3;
    const float sa = img[r00] + img[r00 + 1] + img[r00 + 2];
    const float sb = img[r10] + img[r10 + 1] + img[r10 + 2];
    const float sc = img[r01] + img[r01 + 1] + img[r01 + 2];
    const float sd = img[r11] + img[r11 + 1] + img[r11 + 2];

    out[b * (PATCH * PATCH) + t] =
        (wa * sa + wb * sb + wc * sc + wd * sd) * (1.f / 3.f);
}

extern "C" void kernel_launch(void* const* d_in, const int* in_sizes, int n_in,
                              void* d_out, int out_size, void* d_ws, size_t ws_size,
                              hipStream_t stream) {
    const float* I     = (const float*)d_in[0];
    const float* H_mat = (const float*)d_in[1];
    const float* Mm    = (const float*)d_in[2];
    const float* Mi    = (const float*)d_in[3];
    const int*   pidx  = (const int*)d_in[4];
    // d_in[5] (batch_offsets) is implied by b*IMG_H*IMG_W; unused.
    const int B = in_sizes[1] / 9;

    dim3 grid((PATCH * PATCH) / 256, B, 1);
    warp_gather_kernel<<<grid, 256, 0, stream>>>(I, H_mat, Mm, Mi, pidx,
                                                 (float*)d_out);
    (void)n_in; (void)out_size; (void)d_ws; (void)ws_size;
}